// SumLayer_46059229282760
// MI455X (gfx1250) — compile-verified
//
#include <hip/hip_runtime.h>
#include <stdint.h>

#define BLOCK 256
#define CAP   8192   // ints staged in LDS (32 KB) -> avg need is ~4K ints

__global__ __launch_bounds__(BLOCK) void seg_lse_kernel(
    const float* __restrict__ x,
    const int* __restrict__ ptrs,
    const long long* __restrict__ csr,
    float* __restrict__ out,
    int nseg)
{
    __shared__ int sptrs[CAP];

    const int tid     = threadIdx.x;
    const int segBase = blockIdx.x * BLOCK;
    const int segLast = min(segBase + BLOCK, nseg);

    // Block-wide contiguous edge window [e0, eEnd)
    const long long e0   = csr[segBase];
    const long long eEnd = csr[segLast];

    // Round window start down to a 16B boundary for b128 async staging.
    const long long a0         = e0 & ~3LL;
    const int       totalInts  = (int)(eEnd - a0);
    const int       nChunks    = min((totalInts + 3) >> 2, CAP >> 2);
    const int       stagedInts = nChunks << 2;
    const int* gbase = ptrs + a0;

    // CDNA5 async vector-memory -> LDS copy (tracked by ASYNCcnt).
    // VDST = LDS byte address (low 32 bits of generic shared pointer),
    // VADDR = 64-bit global address pair, SADDR = off.
    for (int c = tid; c < nChunks; c += BLOCK) {
        unsigned           ldsAddr = (unsigned)(uintptr_t)(&sptrs[c << 2]);
        unsigned long long gAddr   = (unsigned long long)(uintptr_t)(gbase + (c << 2));
        asm volatile("global_load_async_to_lds_b128 %0, %1, off"
                     :: "v"(ldsAddr), "v"(gAddr)
                     : "memory");
    }
#if __has_builtin(__builtin_amdgcn_s_wait_asynccnt)
    __builtin_amdgcn_s_wait_asynccnt(0);
#else
    asm volatile("s_wait_asynccnt 0" ::: "memory");
#endif
    __syncthreads();

    const int s = segBase + tid;
    if (s < nseg) {
        const long long b = csr[s];
        const long long e = csr[s + 1];
        const int lb = (int)(b - a0);
        const int le = (int)(e - a0);

        // Pass 1: segment max; rewrite own LDS slots with gathered values
        // so pass 2 avoids a second random global gather.
        float m = -__builtin_inff();
        for (int j = lb; j < le; ++j) {
            float v;
            if (j < stagedInts) {
                const int idx = sptrs[j];
                v = x[idx];
                sptrs[j] = __float_as_int(v);
            } else {
                v = x[ptrs[a0 + (long long)j]];
            }
            m = fmaxf(m, v);
        }

        // Pass 2: sum of exp(v - m), values now resident in LDS.
        float sum = 0.0f;
        for (int j = lb; j < le; ++j) {
            const float v = (j < stagedInts) ? __int_as_float(sptrs[j])
                                             : x[ptrs[a0 + (long long)j]];
            sum += __expf(v - m);
        }

        out[s] = __logf(sum + 1e-15f) + m;
    }
}

extern "C" void kernel_launch(void* const* d_in, const int* in_sizes, int n_in,
                              void* d_out, int out_size, void* d_ws, size_t ws_size,
                              hipStream_t stream) {
    const float*     x    = (const float*)d_in[0];
    const int*       ptrs = (const int*)d_in[1];
    const long long* csr  = (const long long*)d_in[2];
    float*           out  = (float*)d_out;

    const int nseg = in_sizes[2] - 1;   // len(csr) - 1
    const int grid = (nseg + BLOCK - 1) / BLOCK;

    seg_lse_kernel<<<grid, BLOCK, 0, stream>>>(x, ptrs, csr, out, nseg);
}